// DelayTLoss_3925600108635
// MI455X (gfx1250) — compile-verified
//
#include <hip/hip_runtime.h>
#include <math.h>

#define NEGV  (-1.0e30f)
#define Bn    16
#define TmD   256
#define Uu    64
#define U1    65
#define Vv    512
#define KP    68     // K padded to 17*4
#define ROWS_PER_CHUNK 8
#define CHUNK_ELEMS (ROWS_PER_CHUNK * Vv)   // 4096 floats = 16KB

typedef __attribute__((ext_vector_type(2))) float v2f;
typedef __attribute__((ext_vector_type(8))) float v8f;
typedef __attribute__((ext_vector_type(4))) unsigned int su4;
typedef __attribute__((ext_vector_type(8))) unsigned int su8;

// ---------------- TDM: 1-D tile DMA, global -> LDS ----------------
// D# per CDNA5 ISA ch.8: group0 = {count/flags, lds_addr, gaddr_lo, gaddr_hi|type},
// group1 = {mask/data_size, dims/strides}. 1-D tile: tile_dim0 = nelem (<=65535).
__device__ __forceinline__ void tdm_load_f32(unsigned lds_byte_off,
                                             const float* gsrc,
                                             unsigned nelem) {
#if defined(__HIP_DEVICE_COMPILE__)
  unsigned long long ga = (unsigned long long)(size_t)gsrc;
  su4 g0;
  g0[0] = 1u;                                   // count=1, user mode, no gather
  g0[1] = lds_byte_off;                         // LDS byte address
  g0[2] = (unsigned)(ga & 0xffffffffu);         // global_addr[31:0]
  g0[3] = (unsigned)(ga >> 32) | (2u << 30);    // global_addr[56:32] | type=2
  su8 g1;
  g1[0] = (2u << 16);                           // workgroup_mask=0, data_size=4B
  g1[1] = (nelem & 0xffffu) << 16;              // tensor_dim0[15:0]
  g1[2] = (nelem >> 16) | (1u << 16);           // tensor_dim0[31:16], tensor_dim1=1
  g1[3] = (nelem & 0xffffu) << 16;              // tile_dim0 = nelem
  g1[4] = 1u;                                   // tile_dim1=1, tile_dim2=0
  g1[5] = nelem;                                // tensor_dim0_stride[31:0]
  g1[6] = 0u;
  g1[7] = 0u;
  asm volatile("tensor_load_to_lds %0, %1" :: "s"(g0), "s"(g1) : "memory");
#else
  (void)lds_byte_off; (void)gsrc; (void)nelem;
#endif
}

__device__ __forceinline__ void wait_tensor_le(int n) {
#if defined(__HIP_DEVICE_COMPILE__)
  if (n == 0) __builtin_amdgcn_s_wait_tensorcnt(0);
  else        __builtin_amdgcn_s_wait_tensorcnt(1);
#else
  (void)n;
#endif
}

__device__ __forceinline__ v8f wmma4_f32(v2f a, v2f b, v8f c) {
#if defined(__HIP_DEVICE_COMPILE__)
  return __builtin_amdgcn_wmma_f32_16x16x4_f32(false, a, false, b, (short)0, c,
                                               false, false);
#else
  return c;
#endif
}

__device__ __forceinline__ float logaddexpf_(float a, float b) {
  float mx = fmaxf(a, b);
  float mn = fminf(a, b);
  return mx + log1pf(__expf(mn - mx));
}

// ---------------- K1: streaming log-softmax gather (TDM double-buffered) ----
// One wave per 512-logit row; chunk = 8 rows = 16KB per TDM tile.
__global__ __launch_bounds__(256) void k_stats(const float* __restrict__ acts,
                                               const int* __restrict__ labels,
                                               float* __restrict__ blank_lp,
                                               float* __restrict__ label_lp,
                                               int chunks_per_wg) {
  __shared__ float lds[2][ROWS_PER_CHUNK][Vv];   // 32 KB
  const int tid  = threadIdx.x;
  const int wave = tid >> 5;
  const int lane = tid & 31;
  const long long chunk0 = (long long)blockIdx.x * chunks_per_wg;

  if (wave == 0) {
    tdm_load_f32(0u, acts + chunk0 * CHUNK_ELEMS, CHUNK_ELEMS);
  }
  for (int i = 0; i < chunks_per_wg; ++i) {
    if (wave == 0) {
      if (i + 1 < chunks_per_wg) {
        tdm_load_f32(((i + 1) & 1) ? (unsigned)(ROWS_PER_CHUNK * Vv * 4) : 0u,
                     acts + (chunk0 + i + 1) * CHUNK_ELEMS, CHUNK_ELEMS);
        wait_tensor_le(1);   // chunk i has landed (in-order completion)
      } else {
        wait_tensor_le(0);
      }
    }
    __syncthreads();

    const float* row = &lds[i & 1][wave][0];
    float v[16];
#pragma unroll
    for (int j = 0; j < 4; ++j) {
      const float4 q = ((const float4*)row)[lane + 32 * j];
      v[4 * j + 0] = q.x; v[4 * j + 1] = q.y;
      v[4 * j + 2] = q.z; v[4 * j + 3] = q.w;
    }
    float m = v[0];
#pragma unroll
    for (int e = 1; e < 16; ++e) m = fmaxf(m, v[e]);
#pragma unroll
    for (int off = 16; off >= 1; off >>= 1) m = fmaxf(m, __shfl_xor(m, off, 32));
    float s = 0.f;
#pragma unroll
    for (int e = 0; e < 16; ++e) s += __expf(v[e] - m);
#pragma unroll
    for (int off = 16; off >= 1; off >>= 1) s += __shfl_xor(s, off, 32);
    const float lse = m + __logf(s);

    if (lane == 0) {
      const long long r = (chunk0 + i) * ROWS_PER_CHUNK + wave;
      const int b   = (int)(r / (TmD * U1));
      const int rem = (int)(r % (TmD * U1));
      const int t = rem / U1;
      const int u = rem % U1;
      blank_lp[r] = row[0] - lse;                 // x[BLANK=0] - lse
      if (u < Uu) {
        const int l = labels[b * Uu + u];
        label_lp[(b * TmD + t) * Uu + u] = row[l] - lse;
      }
    }
    __syncthreads();   // protect buf[i&1] before next overwrite
  }
}

// ---------------- K2: exclusive cumsum of label_lp along u -> Lc ------------
__global__ void k_cumsum(const float* __restrict__ label_lp,
                         float* __restrict__ Lc) {
  int row = blockIdx.x * blockDim.x + threadIdx.x;   // row = b*Tm + t
  if (row >= Bn * TmD) return;
  float s = 0.f;
  Lc[row * U1] = 0.f;
  for (int u = 0; u < Uu; ++u) {
    s += label_lp[row * Uu + u];
    Lc[row * U1 + u + 1] = s;
  }
}

// ---------------- K3: alpha (block 0) / beta (block 1) DP -------------------
// Per t-step: X(16x65) -> rowmax M -> E=exp(X-M) -> P = E x TriOnes via
// V_WMMA_F32_16X16X4_F32 (M=16 batches, 5 N-tiles x 17 K-chunks) -> log.
__global__ __launch_bounds__(160) void k_dp(const float* __restrict__ blank_lp,
                                            const float* __restrict__ Lc,
                                            const int* __restrict__ act_lens,
                                            const int* __restrict__ label_lens,
                                            float* __restrict__ alpha,
                                            float* __restrict__ beta) {
  __shared__ float Xs[Bn][KP];
  __shared__ float Prev[Bn][U1];
  __shared__ float Mrow[Bn];
  __shared__ int   s_al[Bn], s_ll[Bn];

  const int tid  = threadIdx.x;
  const int lane = tid & 31;
  const int wave = tid >> 5;       // 5 waves: one N-tile each
  const int half = lane >> 4;
  const int mr   = lane & 15;
  const bool fwd = (blockIdx.x == 0);

  if (tid < Bn) { s_al[tid] = act_lens[tid]; s_ll[tid] = label_lens[tid]; }
  for (int idx = tid; idx < Bn * U1; idx += 160)
    Prev[idx / U1][idx % U1] = NEGV;
  __syncthreads();

  for (int step = 0; step < TmD; ++step) {
    const int t = fwd ? step : (TmD - 1 - step);

    // 1. build X
    for (int idx = tid; idx < Bn * KP; idx += 160) {
      const int b = idx / KP, u = idx % KP;
      float x;
      if (u >= U1) {
        x = NEGV;                                   // pad -> E ~ 0
      } else if (fwd) {
        const float lc = Lc[(b * TmD + t) * U1 + u];
        if (t == 0) x = (u == 0 ? 0.f : NEGV) - lc;
        else        x = Prev[b][u] + blank_lp[(b * TmD + (t - 1)) * U1 + u] - lc;
      } else {
        const float bl = blank_lp[(b * TmD + t) * U1 + u];
        const float ex = ((t == s_al[b] - 1) && (u == s_ll[b])) ? bl : NEGV;
        x = logaddexpf_(bl + Prev[b][u], ex) + Lc[(b * TmD + t) * U1 + u];
      }
      Xs[b][u] = x;
    }
    __syncthreads();

    // 2. row max
    if (tid < Bn) {
      float m = Xs[tid][0];
      for (int u = 1; u < U1; ++u) m = fmaxf(m, Xs[tid][u]);
      Mrow[tid] = m;
    }
    __syncthreads();

    // 3. E = exp(X - M)
    for (int idx = tid; idx < Bn * KP; idx += 160) {
      const int b = idx / KP, u = idx % KP;
      Xs[b][u] = __expf(Xs[b][u] - Mrow[b]);
    }
    __syncthreads();

    // 4. prefix-sum-as-matmul: P = E x L (L triangular ones), one N-tile/wave
    const int n0 = wave * 16;
    const int n  = n0 + mr;
    v8f c = {0.f, 0.f, 0.f, 0.f, 0.f, 0.f, 0.f, 0.f};
#pragma unroll
    for (int kc = 0; kc < 17; ++kc) {
      const int kb = kc * 4;
      const int k0 = kb + 2 * half;
      v2f a, bmat;
      a.x = Xs[mr][k0 + 0];
      a.y = Xs[mr][k0 + 1];
      if (fwd) {                       // inclusive forward prefix: k <= n
        bmat.x = (k0     <= n) ? 1.f : 0.f;
        bmat.y = (k0 + 1 <= n) ? 1.f : 0.f;
      } else {                         // inclusive reverse prefix: k >= n
        bmat.x = (k0     >= n) ? 1.f : 0.f;
        bmat.y = (k0 + 1 >= n) ? 1.f : 0.f;
      }
      c = wmma4_f32(a, bmat, c);
    }

    // 5. write back: lane holds D rows M=r+8*half at col n
#pragma unroll
    for (int r = 0; r < 8; ++r) {
      const int b = r + 8 * half;
      const int u = n;
      if (u < U1) {
        const float lc = Lc[(b * TmD + t) * U1 + u];
        float val = Mrow[b] + __logf(fmaxf(c[r], 1e-37f));
        val = fwd ? (lc + val) : (val - lc);
        Prev[b][u] = val;
        (fwd ? alpha : beta)[(b * TmD + t) * U1 + u] = val;
      }
    }
    __syncthreads();
  }
}

// ---------------- K4: logZ per batch ----------------------------------------
__global__ void k_logz(const float* __restrict__ alpha,
                       const float* __restrict__ blank_lp,
                       const int* __restrict__ act_lens,
                       const int* __restrict__ label_lens,
                       float* __restrict__ logZ) {
  int b = threadIdx.x;
  if (b < Bn) {
    int t = act_lens[b] - 1, u = label_lens[b];
    logZ[b] = alpha[(b * TmD + t) * U1 + u] + blank_lp[(b * TmD + t) * U1 + u];
  }
}

// ---------------- K5: deterministic delay reduction (one block per b) -------
__global__ __launch_bounds__(256) void k_delay(const float* __restrict__ alpha,
                                               const float* __restrict__ beta,
                                               const float* __restrict__ label_lp,
                                               const float* __restrict__ logZ,
                                               const int* __restrict__ act_lens,
                                               float* __restrict__ partial) {
  __shared__ float red[256];
  const int b = blockIdx.x;
  const float lz = logZ[b];
  const float inv_len = 1.f / (float)act_lens[b];
  float acc = 0.f;
  for (int idx = threadIdx.x; idx < TmD * Uu; idx += 256) {
    const int t = idx / Uu, u = idx % Uu;
    const float g = __expf(alpha[(b * TmD + t) * U1 + u] +
                           label_lp[(b * TmD + t) * Uu + u] +
                           beta[(b * TmD + t) * U1 + u + 1] - lz);
    acc += g * ((float)t * inv_len);
  }
  red[threadIdx.x] = acc;
  __syncthreads();
  for (int s = 128; s > 0; s >>= 1) {
    if (threadIdx.x < s) red[threadIdx.x] += red[threadIdx.x + s];
    __syncthreads();
  }
  if (threadIdx.x == 0) partial[b] = red[0];
}

__global__ void k_final(const float* __restrict__ logZ,
                        const float* __restrict__ partial,
                        float* __restrict__ out) {
  if (threadIdx.x == 0) {
    float rn = 0.f, dl = 0.f;
    for (int b = 0; b < Bn; ++b) { rn += -logZ[b]; dl += partial[b]; }
    out[0] = rn + dl;   // loss_total  (DELAY_SCALE = 1)
    out[1] = rn;        // loss_rnnt
    out[2] = dl;        // loss_delay
  }
}

extern "C" void kernel_launch(void* const* d_in, const int* in_sizes, int n_in,
                              void* d_out, int out_size, void* d_ws,
                              size_t ws_size, hipStream_t stream) {
  (void)in_sizes; (void)n_in; (void)out_size; (void)ws_size;
  const float* acts       = (const float*)d_in[0];
  const int*   labels     = (const int*)d_in[1];
  const int*   act_lens   = (const int*)d_in[2];
  const int*   label_lens = (const int*)d_in[3];

  float* ws       = (float*)d_ws;
  float* blank_lp = ws;                               // 16*256*65
  float* label_lp = blank_lp + Bn * TmD * U1;         // 16*256*64
  float* Lc       = label_lp + Bn * TmD * Uu;         // 16*256*65
  float* alpha    = Lc       + Bn * TmD * U1;         // 16*256*65
  float* beta     = alpha    + Bn * TmD * U1;         // 16*256*65
  float* logZ     = beta     + Bn * TmD * U1;         // 16
  float* partial  = logZ + Bn;                        // 16

  // 266240 rows / 8 rows-per-chunk = 33280 chunks = 1040 WGs * 32 chunks
  k_stats <<<1040, 256, 0, stream>>>(acts, labels, blank_lp, label_lp, 32);
  k_cumsum<<<(Bn * TmD + 255) / 256, 256, 0, stream>>>(label_lp, Lc);
  k_dp    <<<2, 160, 0, stream>>>(blank_lp, Lc, act_lens, label_lens, alpha, beta);
  k_logz  <<<1, 32, 0, stream>>>(alpha, blank_lp, act_lens, label_lens, logZ);
  k_delay <<<Bn, 256, 0, stream>>>(alpha, beta, label_lp, logZ, act_lens, partial);
  k_final <<<1, 32, 0, stream>>>(logZ, partial, (float*)d_out);
}